// Head_17686675325184
// MI455X (gfx1250) — compile-verified
//
#include <hip/hip_runtime.h>
#include <math.h>
#include <stdint.h>

// Problem constants (match reference setup_inputs)
constexpr int BATCH = 64;
constexpr int NSLOTS = 4096;
constexpr int DDIM = 256;

typedef float v2f __attribute__((ext_vector_type(2)));
typedef float v8f __attribute__((ext_vector_type(8)));

// ---------------- Kernel 1: content addressing logits = beta * cosine ----------------
// Grid: (NSLOTS/128, BATCH), Block: 256 threads (8 waves). Each wave owns 16 rows.
// m is streamed once via GLOBAL_LOAD_ASYNC_TO_LDS_B128 (memory->LDS DMA, ASYNCcnt)
// into double-buffered LDS slabs (row stride 36 floats, conflict-free fragment reads),
// then V_WMMA_F32_16X16X4_F32 accumulates the 16 row-dots against k; row norms are
// accumulated on the VALU from the same staged data (co-executes with XDL WMMA).

constexpr int ROWS_PER_BLOCK = 128;
constexpr int K1_THREADS = 256;
constexpr int K1_WAVES = K1_THREADS / 32;
constexpr int CHUNK = 32;            // D-columns staged per slab
constexpr int NCHUNK = DDIM / CHUNK; // 8
constexpr int LDS_STRIDE = 36;       // padded row stride in floats (144B, 16B aligned)
constexpr int SLAB = ROWS_PER_BLOCK * LDS_STRIDE;

__global__ __launch_bounds__(K1_THREADS)
void ntm_content_kernel(const float* __restrict__ m,
                        const float* __restrict__ k,
                        const float* __restrict__ beta,
                        float* __restrict__ logits)
{
    __shared__ float tile[2 * SLAB];                      // 2 x 18432 B (double buffer)
    __shared__ float k_s[DDIM];                           // 1 KB
    __shared__ float red_s[K1_WAVES];
    __shared__ float dots_s[ROWS_PER_BLOCK];
    __shared__ float norms_s[ROWS_PER_BLOCK];
    __shared__ float knorm_s;

    const int tid   = threadIdx.x;
    const int lane  = tid & 31;
    const int wid   = tid >> 5;
    const int batch = blockIdx.y;
    const int rowBase = blockIdx.x * ROWS_PER_BLOCK;

    const long mBase = ((long)batch * NSLOTS + rowBase) * DDIM;

    // Per-thread staging coordinates: 256 threads x 16B = 32 rows x 32 cols per slab.
    const int srow = tid >> 3;
    const int scol = (tid & 7) << 2;   // float column within chunk

    // Issue one async 16B memory->LDS DMA for chunk c into buffer buf.
    auto issue_async = [&](int buf, int c) {
        const float* g = m + mBase + (long)srow * DDIM + c * CHUNK + scol;
        const unsigned ldsOff =
            (unsigned)(uintptr_t)(&tile[buf * SLAB + srow * LDS_STRIDE + scol]);
        asm volatile("global_load_async_to_lds_b128 %0, %1, off"
                     :: "v"(ldsOff), "v"(g) : "memory");
    };

    // Prologue: start DMA of chunk 0 into buffer 0 immediately.
    issue_async(0, 0);

    // ---- stage k[batch] and compute ||k|| (block-wide, D == K1_THREADS) ----
    const float kv = k[batch * DDIM + tid];
    k_s[tid] = kv;
    float p = kv * kv;
    #pragma unroll
    for (int off = 16; off > 0; off >>= 1) p += __shfl_xor(p, off, 32);
    if (lane == 0) red_s[wid] = p;
    __syncthreads();
    if (tid == 0) {
        float ssum = 0.f;
        #pragma unroll
        for (int i = 0; i < K1_WAVES; ++i) ssum += red_s[i];
        knorm_s = sqrtf(ssum);
    }

    // ---- main streaming loop: double-buffered async DMA + WMMA ----
    v8f acc = {};                   // C accumulator: 16 row-dots (all N cols equal)
    float norm2p = 0.f;             // per-lane partial sum of squares
    const int waveRow = wid * 16;   // wave's first row within the 128-row block tile
    const int half = lane >> 4;     // 0: K%4 in {0,1}; 1: K%4 in {2,3}
    const int lrow = lane & 15;     // row within wave tile

    #pragma unroll
    for (int c = 0; c < NCHUNK; ++c) {
        if (c + 1 < NCHUNK) {
            // Kick next chunk's DMA into the other buffer, then wait for current
            // chunk only (1 DMA may remain in flight => overlap with compute).
            issue_async((c + 1) & 1, c + 1);
            asm volatile("s_wait_asynccnt 0x1" ::: "memory");
        } else {
            asm volatile("s_wait_asynccnt 0x0" ::: "memory");
        }
        __syncthreads();            // all waves' slab-c DMA visible block-wide

        const float* slab = &tile[(c & 1) * SLAB];
        #pragma unroll
        for (int j = 0; j < CHUNK / 4; ++j) {
            const int kk   = c * CHUNK + j * 4;        // global K base
            const int lcol = j * 4 + half * 2;         // local col for this lane's A frag
            const float* ap = &slab[(waveRow + lrow) * LDS_STRIDE + lcol];
            v2f a;  a.x = ap[0];               a.y = ap[1];
            v2f bf; bf.x = k_s[kk + half * 2]; bf.y = k_s[kk + half * 2 + 1];
            // row-norm partials from the exact same staged values (VALU, co-exec w/ WMMA)
            norm2p = fmaf(a.x, a.x, norm2p);
            norm2p = fmaf(a.y, a.y, norm2p);
            // D(16x16) += A(16x4, m rows) x B(4x16, k broadcast to all columns)
            acc = __builtin_amdgcn_wmma_f32_16x16x4_f32(
                      false, a, false, bf, (short)0, acc, false, false);
        }
        __syncthreads();            // slab free for reuse before iteration c+2's DMA
    }

    // ---- fold norms across lane halves; scatter dots from C layout ----
    const float norm2 = norm2p + __shfl_xor(norm2p, 16, 32);
    if (half == 0) norms_s[waveRow + lrow] = norm2;
    // C layout: VGPR r = row r (lanes 0-15) / row r+8 (lanes 16-31); all cols equal
    if (lane == 0) {
        #pragma unroll
        for (int r2 = 0; r2 < 8; ++r2) dots_s[waveRow + r2] = acc[r2];
    }
    if (lane == 16) {
        #pragma unroll
        for (int r2 = 0; r2 < 8; ++r2) dots_s[waveRow + 8 + r2] = acc[r2];
    }
    __syncthreads();

    if (tid < ROWS_PER_BLOCK) {
        const float mn   = fmaxf(sqrtf(norms_s[tid]), 1e-8f);
        const float kn   = fmaxf(knorm_s, 1e-8f);
        const float cosv = dots_s[tid] / (mn * kn);
        logits[(long)batch * NSLOTS + rowBase + tid] = beta[batch] * cosv;
    }
}

// ---------------- Kernel 2: softmax + interpolate + circular shift + sharpen ----------------
constexpr int K2_THREADS = 512;
constexpr int EPT = NSLOTS / K2_THREADS;   // 8 elements per thread

__device__ __forceinline__ float block_reduce_sum(float v, float* red, int tid) {
    const int lane = tid & 31, wid = tid >> 5;
    #pragma unroll
    for (int off = 16; off > 0; off >>= 1) v += __shfl_xor(v, off, 32);
    if (lane == 0) red[wid] = v;
    __syncthreads();
    float t = (lane < (K2_THREADS >> 5)) ? red[lane] : 0.f;
    #pragma unroll
    for (int off = 16; off > 0; off >>= 1) t += __shfl_xor(t, off, 32);
    __syncthreads();                        // red[] reusable afterwards
    return t;                               // uniform across all lanes/waves
}

__device__ __forceinline__ float block_reduce_max(float v, float* red, int tid) {
    const int lane = tid & 31, wid = tid >> 5;
    #pragma unroll
    for (int off = 16; off > 0; off >>= 1) v = fmaxf(v, __shfl_xor(v, off, 32));
    if (lane == 0) red[wid] = v;
    __syncthreads();
    float t = (lane < (K2_THREADS >> 5)) ? red[lane] : -INFINITY;
    #pragma unroll
    for (int off = 16; off > 0; off >>= 1) t = fmaxf(t, __shfl_xor(t, off, 32));
    __syncthreads();
    return t;
}

__global__ __launch_bounds__(K2_THREADS)
void ntm_finalize_kernel(const float* __restrict__ logits,
                         const float* __restrict__ g,
                         const float* __restrict__ s,
                         const float* __restrict__ r,
                         const float* __restrict__ w_prev,
                         float* __restrict__ out)
{
    __shared__ float w_s[NSLOTS];                 // 16 KB
    __shared__ float red_s[K2_THREADS / 32];

    const int tid   = threadIdx.x;
    const int batch = blockIdx.x;
    const float* lg = logits + (long)batch * NSLOTS;

    // 1) stable softmax: max
    float lmax = -INFINITY;
    #pragma unroll
    for (int i = 0; i < EPT; ++i) lmax = fmaxf(lmax, lg[tid + i * K2_THREADS]);
    const float bmax = block_reduce_max(lmax, red_s, tid);

    // 2) exp & sum
    float lsum = 0.f;
    #pragma unroll
    for (int i = 0; i < EPT; ++i) {
        const int n = tid + i * K2_THREADS;
        const float e = expf(lg[n] - bmax);
        w_s[n] = e;
        lsum += e;
    }
    const float Z = block_reduce_sum(lsum, red_s, tid);

    // 3) interpolate:  w = softmax + (1-g) * w_prev   (matches reference exactly)
    const float ginv = 1.0f - g[batch];
    const float invZ = 1.0f / Z;
    #pragma unroll
    for (int i = 0; i < EPT; ++i) {
        const int n = tid + i * K2_THREADS;
        w_s[n] = w_s[n] * invZ + ginv * w_prev[(long)batch * NSLOTS + n];
    }
    __syncthreads();

    // 4) circular shift: w'[n] = s0*w[n-1] + s1*w[n] + s2*w[n+1]  (mod N)
    const float s0 = s[batch * 3 + 0];
    const float s1 = s[batch * 3 + 1];
    const float s2 = s[batch * 3 + 2];
    float wn[EPT];
    #pragma unroll
    for (int i = 0; i < EPT; ++i) {
        const int n  = tid + i * K2_THREADS;
        const int nm = (n == 0) ? NSLOTS - 1 : n - 1;
        const int np = (n == NSLOTS - 1) ? 0 : n + 1;
        wn[i] = s0 * w_s[nm] + s1 * w_s[n] + s2 * w_s[np];
    }

    // 5) sharpen: w^r / (sum + 1e-16)
    const float rr = r[batch];
    float psum = 0.f;
    #pragma unroll
    for (int i = 0; i < EPT; ++i) {
        wn[i] = powf(wn[i], rr);
        psum += wn[i];
    }
    __syncthreads();                               // red_s reuse barrier
    const float S = block_reduce_sum(psum, red_s, tid);
    const float inv = 1.0f / (S + 1e-16f);
    #pragma unroll
    for (int i = 0; i < EPT; ++i)
        out[(long)batch * NSLOTS + tid + i * K2_THREADS] = wn[i] * inv;
}

// ---------------- launch ----------------
extern "C" void kernel_launch(void* const* d_in, const int* in_sizes, int n_in,
                              void* d_out, int out_size, void* d_ws, size_t ws_size,
                              hipStream_t stream) {
    const float* m      = (const float*)d_in[0];
    const float* k      = (const float*)d_in[1];
    const float* b      = (const float*)d_in[2];
    const float* g      = (const float*)d_in[3];
    const float* s      = (const float*)d_in[4];
    const float* r      = (const float*)d_in[5];
    const float* w_prev = (const float*)d_in[6];
    float* out    = (float*)d_out;
    float* logits = (float*)d_ws;                 // B*N floats = 1 MB scratch

    dim3 grid1(NSLOTS / ROWS_PER_BLOCK, BATCH);   // (32, 64)
    ntm_content_kernel<<<grid1, K1_THREADS, 0, stream>>>(m, k, b, logits);
    ntm_finalize_kernel<<<BATCH, K2_THREADS, 0, stream>>>(logits, g, s, r, w_prev, out);
}